// cheat_PHNN_31404800869102
// MI455X (gfx1250) — compile-verified
//
#include <hip/hip_runtime.h>
#include <math.h>

#define DTS 0.01f
#define SIG 64
#define NIN 2048
#define HID 64

typedef float v2f __attribute__((ext_vector_type(2)));
typedef float v8f __attribute__((ext_vector_type(8)));

// Workgroup barrier that only waits for LDS (DScnt), not global store acks.
// __syncthreads() would insert s_wait_storecnt 0x0 and serialize each step on
// HBM write-ack of the y stores, which nobody ever re-reads.
__device__ __forceinline__ void wg_lds_barrier() {
  asm volatile(
      "s_wait_dscnt 0x0\n\t"
      "s_barrier_signal -1\n\t"
      "s_barrier_wait -1" ::: "memory");
}

// ---------------------------------------------------------------------------
// Prep: build A = JmR@Q, then R = dt(I + dt/2 A(I + dt/3 A(I + dt/4 A))),
// Phi = I + R@A, Bd = R[:,64:], and pack B-operand layouts for f32 WMMA.
//   phi2[k2*128+n] : float2 {Phi[n][2k2], Phi[n][2k2+1]}        (k2<64)
//   bd2 [k2*128+n] : float2 {R[n][64+2k2], R[n][64+2k2+1]}       (k2<32)
//   qt2 [k2*64+np] : float2 {M[np][2k2], M[np][2k2+1]}/4         (k2<32)
// ---------------------------------------------------------------------------
__global__ __launch_bounds__(128) void prep_kernel(
    const float* __restrict__ Mm, const float* __restrict__ Dm, const float* __restrict__ Km,
    float* __restrict__ phi2, float* __restrict__ bd2, float* __restrict__ qt2,
    float* __restrict__ Asc, float* __restrict__ Tsc, float* __restrict__ Rsc,
    float* __restrict__ DMsc)
{
  const int t = threadIdx.x;           // 0..127
  const float h = DTS;
  if (t < 64) {
    for (int j = 0; j < 64; ++j) {
      float s = 0.f;
      for (int k = 0; k < 64; ++k) s += Dm[t*64+k] * Mm[k*64+j];
      DMsc[t*64+j] = 0.25f * s;
    }
  }
  __threadfence(); __syncthreads();
  for (int j = 0; j < 128; ++j) {
    float a;
    if (t < 64) a = (j >= 64) ? 0.25f * Mm[t*64 + (j-64)] : 0.f;
    else        a = (j < 64) ? -Km[(t-64)*64 + j] : -DMsc[(t-64)*64 + (j-64)];
    Asc[t*128+j] = a;
  }
  __threadfence(); __syncthreads();
  for (int j = 0; j < 128; ++j)
    Tsc[t*128+j] = ((t==j)?1.f:0.f) + (h*0.25f)*Asc[t*128+j];
  __threadfence(); __syncthreads();
  for (int j = 0; j < 128; ++j) {
    float s = 0.f;
    for (int k = 0; k < 128; ++k) s += Asc[t*128+k]*Tsc[k*128+j];
    Rsc[t*128+j] = ((t==j)?1.f:0.f) + (h/3.f)*s;
  }
  __threadfence(); __syncthreads();
  for (int j = 0; j < 128; ++j) {
    float s = 0.f;
    for (int k = 0; k < 128; ++k) s += Asc[t*128+k]*Rsc[k*128+j];
    Tsc[t*128+j] = h*(((t==j)?1.f:0.f) + 0.5f*h*s);
  }
  __threadfence(); __syncthreads();
  for (int j = 0; j < 128; ++j) {
    float s = 0.f;
    for (int k = 0; k < 128; ++k) s += Tsc[t*128+k]*Asc[k*128+j];
    Rsc[t*128+j] = ((t==j)?1.f:0.f) + s;
  }
  __threadfence(); __syncthreads();
  for (int p = 0; p < 64; ++p) {      // phi2: k2=p, n=t
    phi2[(p*128+t)*2+0] = Rsc[t*128 + 2*p + 0];
    phi2[(p*128+t)*2+1] = Rsc[t*128 + 2*p + 1];
  }
  for (int p = 0; p < 32; ++p) {      // bd2: k2=p, n=t
    bd2[(p*128+t)*2+0] = Tsc[t*128 + 64 + 2*p + 0];
    bd2[(p*128+t)*2+1] = Tsc[t*128 + 64 + 2*p + 1];
  }
  if (t < 64) {
    for (int p = 0; p < 32; ++p) {    // qt2: k2s=p, np=t
      qt2[(p*64+t)*2+0] = 0.25f*Mm[t*64 + 2*p + 0];
      qt2[(p*64+t)*2+1] = 0.25f*Mm[t*64 + 2*p + 1];
    }
  }
}

// ---------------------------------------------------------------------------
// Encoder: x0 = data@W_lin + b_lin + tanh(tanh(data@W1+b1)@W2+b2)@W3 + b3
// ---------------------------------------------------------------------------
__global__ __launch_bounds__(256) void enc_kernel(
    const float* __restrict__ up, const float* __restrict__ yp,
    const float* __restrict__ Wl, const float* __restrict__ bl,
    const float* __restrict__ W1, const float* __restrict__ b1,
    const float* __restrict__ W2, const float* __restrict__ b2,
    const float* __restrict__ W3, const float* __restrict__ b3,
    float* __restrict__ x0buf)
{
  __shared__ float d[NIN];
  __shared__ float h1s[HID];
  __shared__ float h2s[HID];
  const int b = blockIdx.x, j = threadIdx.x;
  for (int i = j; i < 1024; i += 256) d[i]      = up[(size_t)b*1024 + i];
  for (int i = j; i < 1024; i += 256) d[1024+i] = yp[(size_t)b*1024 + i];
  __syncthreads();
  float lin = 0.f;
  if (j < 128) {
    float s = bl[j];
    for (int k = 0; k < NIN; ++k) s += d[k]*Wl[k*128 + j];
    lin = s;
  } else if (j < 192) {
    const int c = j - 128;
    float s = b1[c];
    for (int k = 0; k < NIN; ++k) s += d[k]*W1[k*64 + c];
    h1s[c] = tanhf(s);
  }
  __syncthreads();
  if (j < 64) {
    float s = b2[j];
    for (int k = 0; k < 64; ++k) s += h1s[k]*W2[k*64+j];
    h2s[j] = tanhf(s);
  }
  __syncthreads();
  if (j < 128) {
    float s = lin + b3[j];
    for (int k = 0; k < 64; ++k) s += h2s[k]*W3[k*128+j];
    x0buf[(size_t)b*128 + j] = s;
  }
}

// ---------------------------------------------------------------------------
// Rollout: per step  y_t = (x Q^T)[:,64:],  x <- x@Phi^T + u_t@Bd^T
// 16 batch rows per WG; 8 waves, one Phi N-tile each (<256 VGPRs). accD split
// into two parity chains for wmma-pipe ILP; y store overlapped with the Bd
// tail; u[t+1] load software-pipelined; ping-pong LDS buffers -> ONE
// dscnt-only barrier per step.
// ---------------------------------------------------------------------------
#define ROLL_STEP(tcur, XR, XW, UR, UW)                                       \
  {                                                                           \
    const int tc = (tcur);                                                    \
    float4 unext = {};                                                        \
    const bool hasnext = tc + 1 < T;                                          \
    if (hasnext) unext = *(const float4*)(uptr + (size_t)(tc + 1)*SIG);       \
    v8f accDa = {}, accDb = {}, accY = {};                                    \
    _Pragma("unroll")                                                         \
    for (int kk = 0; kk < 32; ++kk) {                                         \
      const int k0 = kk*4 + 2*kh;                                             \
      v2f a = *(const v2f*)((XR) + lnm*132 + k0);                             \
      if (kk & 1)                                                             \
        accDb = __builtin_amdgcn_wmma_f32_16x16x4_f32(false, a, false,        \
                    phiB[kk], (short)0, accDb, false, false);                 \
      else                                                                    \
        accDa = __builtin_amdgcn_wmma_f32_16x16x4_f32(false, a, false,        \
                    phiB[kk], (short)0, accDa, false, false);                 \
      if (kk >= 16)  /* Q^T output map only touches K in [64,128) */          \
        accY = __builtin_amdgcn_wmma_f32_16x16x4_f32(false, a, false,         \
                    qtB[kk-16], (short)0, accY, false, false);                \
    }                                                                         \
    if (wave < 4) {                                                           \
      float* ob = ybase + (size_t)tc * SIG;                                   \
      _Pragma("unroll")                                                       \
      for (int i = 0; i < 8; ++i) ob[(size_t)i * T * SIG] = accY[i];          \
    }                                                                         \
    _Pragma("unroll")                                                         \
    for (int kk = 0; kk < 16; ++kk) {                                         \
      const int k0 = kk*4 + 2*kh;                                             \
      v2f a = *(const v2f*)((UR) + lnm*68 + k0);                              \
      if (kk & 1)                                                             \
        accDb = __builtin_amdgcn_wmma_f32_16x16x4_f32(false, a, false,        \
                    bdB[kk], (short)0, accDb, false, false);                  \
      else                                                                    \
        accDa = __builtin_amdgcn_wmma_f32_16x16x4_f32(false, a, false,        \
                    bdB[kk], (short)0, accDa, false, false);                  \
    }                                                                         \
    v8f accD = accDa + accDb;                                                 \
    {                                                                         \
      const int n = wave*16 + lnm;                                            \
      _Pragma("unroll")                                                       \
      for (int i = 0; i < 8; ++i) (XW)[(i + 8*kh)*132 + n] = accD[i];         \
    }                                                                         \
    if (hasnext) *(float4*)((UW) + urow*68 + uc0) = unext;                    \
    wg_lds_barrier();                                                         \
  }

template <int TT>
__global__ __launch_bounds__(256) void rollout_kernel(
    const float* __restrict__ uf, const float* __restrict__ x0buf,
    const float* __restrict__ phi2, const float* __restrict__ bd2,
    const float* __restrict__ qt2, float* __restrict__ out, int Trt)
{
  const int T = (TT > 0) ? TT : Trt;
  __shared__ __align__(16) float xbuf0[16*132], xbuf1[16*132];
  __shared__ __align__(16) float ubuf0[16*68],  ubuf1[16*68];
  const int tid  = threadIdx.x;
  const int wave = tid >> 5, lane = tid & 31;
  const int lnm  = lane & 15, kh = lane >> 4;   // A/B frag half select
  const int b0   = blockIdx.x * 16;

  // ----- load resident B operands -----
  v2f phiB[32], bdB[16], qtB[16];
  {
    const int n = wave*16 + lnm;
    #pragma unroll
    for (int kk = 0; kk < 32; ++kk)
      phiB[kk] = *(const v2f*)(phi2 + (size_t)((kk*2 + kh)*128 + n)*2);
    #pragma unroll
    for (int kk = 0; kk < 16; ++kk)
      bdB[kk] = *(const v2f*)(bd2 + (size_t)((kk*2 + kh)*128 + n)*2);
  }
  if (wave < 4) {
    const int np = wave*16 + lnm;
    #pragma unroll
    for (int kk = 0; kk < 16; ++kk)
      qtB[kk] = *(const v2f*)(qt2 + (size_t)((kk*2 + kh)*64 + np)*2);
  }

  // ----- load x0 tile into LDS -----
  #pragma unroll
  for (int i = 0; i < 8; ++i) {
    const int idx = tid + 256*i;
    const int r = idx >> 7, c = idx & 127;
    xbuf0[r*132 + c] = x0buf[(size_t)(b0 + r)*128 + c];
  }

  // ----- u staging: thread -> (row, 4 cols); preload t=0 -----
  const int urow = tid >> 4;
  const int uc0  = (tid & 15) * 4;
  const float* uptr = uf + ((size_t)(b0 + urow)*T)*SIG + uc0;
  {
    float4 u0 = *(const float4*)uptr;
    *(float4*)(ubuf0 + urow*68 + uc0) = u0;
  }
  __syncthreads();

  // per-lane output base (row = b0 + 8*kh + i, col = wave*16 + lnm)
  float* ybase = out + ((size_t)(b0 + 8*kh)*T)*SIG + wave*16 + lnm;

  for (int t = 0; t < T; t += 2) {
    ROLL_STEP(t, xbuf0, xbuf1, ubuf0, ubuf1);
    if (TT > 0 || t + 1 < T)
      ROLL_STEP(t + 1, xbuf1, xbuf0, ubuf1, ubuf0);
  }
}

// ---------------------------------------------------------------------------
extern "C" void kernel_launch(void* const* d_in, const int* in_sizes, int n_in,
                              void* d_out, int out_size, void* d_ws, size_t ws_size,
                              hipStream_t stream)
{
  const float* u_past   = (const float*)d_in[0];
  const float* y_past   = (const float*)d_in[1];
  const float* u_future = (const float*)d_in[2];
  const float* W_lin = (const float*)d_in[3];
  const float* b_lin = (const float*)d_in[4];
  const float* W1 = (const float*)d_in[5];
  const float* b1 = (const float*)d_in[6];
  const float* W2 = (const float*)d_in[7];
  const float* b2 = (const float*)d_in[8];
  const float* W3 = (const float*)d_in[9];
  const float* b3 = (const float*)d_in[10];
  const float* Mm = (const float*)d_in[11];
  const float* Dm = (const float*)d_in[12];
  const float* Km = (const float*)d_in[13];

  const int B = in_sizes[0] / (16*SIG);     // 2048
  const int T = in_sizes[2] / (B * SIG);    // 512

  float* ws   = (float*)d_ws;
  float* phi2 = ws;                         // 64*128*2  = 16384 f32
  float* bd2  = ws + 16384;                 // 32*128*2  =  8192 f32
  float* qt2  = ws + 24576;                 // 32*64*2   =  4096 f32
  float* x0b  = ws + 28672;                 // B*128
  float* Asc  = x0b + (size_t)B*128;        // 128*128
  float* Tsc  = Asc + 16384;
  float* Rsc  = Tsc + 16384;
  float* DMsc = Rsc + 16384;                // 64*64

  prep_kernel<<<1, 128, 0, stream>>>(Mm, Dm, Km, phi2, bd2, qt2, Asc, Tsc, Rsc, DMsc);
  enc_kernel<<<B, 256, 0, stream>>>(u_past, y_past, W_lin, b_lin, W1, b1,
                                    W2, b2, W3, b3, x0b);
  if (T == 512)
    rollout_kernel<512><<<B/16, 256, 0, stream>>>(u_future, x0b, phi2, bd2, qt2,
                                                  (float*)d_out, T);
  else
    rollout_kernel<0><<<B/16, 256, 0, stream>>>(u_future, x0b, phi2, bd2, qt2,
                                                (float*)d_out, T);
}